// GraphSAGEModelV1_66176856097277
// MI455X (gfx1250) — compile-verified
//
#include <hip/hip_runtime.h>
#include <hip/hip_bf16.h>

#define N_NODES 50000
#define N_EDGES 1600000
#define D_IN    128
#define D_HID   128
#define D_OUT   64

typedef float v2f __attribute__((ext_vector_type(2)));
typedef float v8f __attribute__((ext_vector_type(8)));

// ---------------------------------------------------------------------------
// Zero-fill workspace regions (graph-capture safe; no hipMemset)
// ---------------------------------------------------------------------------
__global__ void zero_f32(float* __restrict__ p, int n) {
    int i = blockIdx.x * blockDim.x + threadIdx.x;
    if (i < n) p[i] = 0.0f;
}

// ---------------------------------------------------------------------------
// Degree count: cnt[dst[e]] += 1
// ---------------------------------------------------------------------------
__global__ void count_deg(const int* __restrict__ dst, float* __restrict__ cnt, int n_edges) {
    int e = blockIdx.x * blockDim.x + threadIdx.x;
    if (e < n_edges) atomicAdd(&cnt[dst[e]], 1.0f);
}

// ---------------------------------------------------------------------------
// Scatter-add of edge messages: agg[dst[e]][:] += X[src[e]][:]
// One thread handles a 4-float chunk of one edge message.
// ---------------------------------------------------------------------------
__global__ void scatter_add(const float* __restrict__ X,
                            const int* __restrict__ src,
                            const int* __restrict__ dst,
                            float* __restrict__ agg,
                            int n_edges, int d) {
    const int chunks = d >> 2;  // 32 for d=128
    int tid = blockIdx.x * blockDim.x + threadIdx.x;
    int total = n_edges * chunks;
    if (tid >= total) return;
    int e = tid / chunks;
    int c = (tid - e * chunks) << 2;
    int s = src[e];
    int t = dst[e];
    const float4 v = *(const float4*)(X + (size_t)s * d + c);
    float* o = agg + (size_t)t * d + c;
    atomicAdd(o + 0, v.x);
    atomicAdd(o + 1, v.y);
    atomicAdd(o + 2, v.z);
    atomicAdd(o + 3, v.w);
}

// ---------------------------------------------------------------------------
// Fused SAGE layer GEMM via fp32 WMMA (V_WMMA_F32_16X16X4_F32):
//   Out = act( (agg * 1/max(cnt,1)) @ Wl^T + X @ Wr^T + bias )
// One wave (32 lanes) computes one 16x16 output tile.
//   A (16x4 f32): lane = m(0..15) | half(k-pair);  a[r] = A[m][2*half + r]
//   B (4x16 f32): lane = n(0..15) | half(k-pair);  b[r] = B[2*half + r][n] = W[n][2*half+r]
//   C/D (16x16):  reg r, lane L -> M = r + 8*(L>>4), N = L&15
// ---------------------------------------------------------------------------
template <bool RELU>
__global__ __launch_bounds__(32)
void sage_gemm_wmma(const float* __restrict__ Agg,   // [n_rows, d_in] neighbor sums
                    const float* __restrict__ Cnt,   // [n_rows] in-degree
                    const float* __restrict__ X,     // [n_rows, d_in] self features
                    const float* __restrict__ Wl,    // [d_out, d_in] row-major
                    const float* __restrict__ Wr,    // [d_out, d_in] row-major
                    const float* __restrict__ Bias,  // [d_out]
                    float* __restrict__ Out,         // [n_rows, d_out]
                    int d_in, int d_out) {
    const int lane = threadIdx.x;       // 0..31, full wave, no divergence
    const int m    = lane & 15;         // A-row within tile / B-col within tile
    const int half = lane >> 4;         // selects K pair {0,1} vs {2,3}
    const int row0 = blockIdx.x << 4;   // output row tile base
    const int col0 = blockIdx.y << 4;   // output col tile base

    const int arow = row0 + m;          // row of A this lane streams
    const int wrow = col0 + m;          // weight row (output feature) this lane streams

    const float inv = 1.0f / fmaxf(Cnt[arow], 1.0f);  // fold "mean" into A

    const float* aggP = Agg + (size_t)arow * d_in + 2 * half;
    const float* xP   = X   + (size_t)arow * d_in + 2 * half;
    const float* wlP  = Wl  + (size_t)wrow * d_in + 2 * half;
    const float* wrP  = Wr  + (size_t)wrow * d_in + 2 * half;

    v8f acc = {};
    for (int k0 = 0; k0 < d_in; k0 += 4) {
        v2f aMean = *(const v2f*)(aggP + k0);
        aMean *= inv;
        v2f aSelf = *(const v2f*)(xP + k0);
        v2f bL    = *(const v2f*)(wlP + k0);
        v2f bR    = *(const v2f*)(wrP + k0);
        // D = A*B + C, fp32 in / fp32 accumulate
        acc = __builtin_amdgcn_wmma_f32_16x16x4_f32(
                  false, aMean, false, bL, (short)0, acc, false, false);
        acc = __builtin_amdgcn_wmma_f32_16x16x4_f32(
                  false, aSelf, false, bR, (short)0, acc, false, false);
    }

    const float bval = Bias[col0 + m];  // N = m for this lane
#pragma unroll
    for (int r = 0; r < 8; ++r) {
        int orow = row0 + r + 8 * half;
        float v = acc[r] + bval;
        if (RELU) v = fmaxf(v, 0.0f);
        Out[(size_t)orow * d_out + col0 + m] = v;
    }
}

// ---------------------------------------------------------------------------
// Launch
// ---------------------------------------------------------------------------
extern "C" void kernel_launch(void* const* d_in, const int* in_sizes, int n_in,
                              void* d_out, int out_size, void* d_ws, size_t ws_size,
                              hipStream_t stream) {
    const float* x    = (const float*)d_in[0];                 // [N, 128]
    const int*   eidx = (const int*)d_in[1];                   // [2, E]
    const float* Wl1  = (const float*)d_in[2];                 // [128, 128]
    const float* Wr1  = (const float*)d_in[3];                 // [128, 128]
    const float* b1   = (const float*)d_in[4];                 // [128]
    const float* Wl2  = (const float*)d_in[5];                 // [64, 128]
    const float* Wr2  = (const float*)d_in[6];                 // [64, 128]
    const float* b2   = (const float*)d_in[7];                 // [64]
    float*       out  = (float*)d_out;                         // [N, 64]

    const int* src = eidx;                // edge_index[0]
    const int* dst = eidx + N_EDGES;      // edge_index[1]

    // Workspace layout (floats), 256B-aligned segments
    float* ws = (float*)d_ws;
    size_t off = 0;
    float* cnt = ws + off;  off += ((size_t)N_NODES + 63) & ~(size_t)63;          // 50048
    float* agg = ws + off;  off += (size_t)N_NODES * D_IN;                        // 6.4M
    float* h   = ws + off;  off += (size_t)N_NODES * D_HID;                       // 6.4M
    (void)ws_size; (void)in_sizes; (void)n_in; (void)out_size;

    const int nAgg = N_NODES * D_IN;
    const int scatterThreads = N_EDGES * (D_IN / 4);

    dim3 blk256(256);

    // ---- Layer 1 ----
    zero_f32<<<dim3((N_NODES + 255) / 256), blk256, 0, stream>>>(cnt, N_NODES);
    zero_f32<<<dim3((nAgg + 255) / 256), blk256, 0, stream>>>(agg, nAgg);
    count_deg<<<dim3((N_EDGES + 255) / 256), blk256, 0, stream>>>(dst, cnt, N_EDGES);
    scatter_add<<<dim3((scatterThreads + 255) / 256), blk256, 0, stream>>>(
        x, src, dst, agg, N_EDGES, D_IN);
    sage_gemm_wmma<true><<<dim3(N_NODES / 16, D_HID / 16), dim3(32), 0, stream>>>(
        agg, cnt, x, Wl1, Wr1, b1, h, D_IN, D_HID);

    // ---- Layer 2 ----
    zero_f32<<<dim3((nAgg + 255) / 256), blk256, 0, stream>>>(agg, nAgg);
    scatter_add<<<dim3((scatterThreads + 255) / 256), blk256, 0, stream>>>(
        h, src, dst, agg, N_EDGES, D_HID);
    sage_gemm_wmma<false><<<dim3(N_NODES / 16, D_OUT / 16), dim3(32), 0, stream>>>(
        agg, cnt, h, Wl2, Wr2, b2, out, D_HID, D_OUT);
}